// DDI_57260503990391
// MI455X (gfx1250) — compile-verified
//
#include <hip/hip_runtime.h>
#include <cstdint>
#include <cstddef>

// Problem constants (match reference exactly).
#define BATCH   128
#define SEQLEN  3000
#define FDIM    64
#define PATCH   3
#define NCHUNK  ((SEQLEN - PATCH) / PATCH)   // 999
#define NSTAGE  20                           // 3*20 = 60 outstanding <= 63 (ASYNCcnt is 6 bits)
#define HALF    32                           // one wave32 handles 32 f-columns

#define ROW_B   (FDIM * 4)                   // 256 B: row stride in BOTH global and LDS
#define SLOT_B  (PATCH * ROW_B)              // 768 B per pipeline slot
#define SLOT_F  (PATCH * FDIM)               // 192 floats per slot

// ---- CDNA5 async copy: global -> LDS, tracked by ASYNCcnt ------------------
// INST_OFFSET is added to BOTH the LDS and the global address (ISA 08 §4.4),
// so with matching 256B row strides one base pair covers all 3 rows.
template <int OFF>
__device__ __forceinline__ void async_copy_b32_off(uint32_t lds_addr, uint64_t gaddr) {
  asm volatile("global_load_async_to_lds_b32 %0, %1, off offset:%2"
               :: "v"(lds_addr), "v"(gaddr), "n"(OFF)
               : "memory");
}

__device__ __forceinline__ void async_stage(uint32_t lds_addr, uint64_t gaddr) {
  async_copy_b32_off<0 * ROW_B>(lds_addr, gaddr);
  async_copy_b32_off<1 * ROW_B>(lds_addr, gaddr);
  async_copy_b32_off<2 * ROW_B>(lds_addr, gaddr);
}

template <int N>
__device__ __forceinline__ void wait_asynccnt_le() {
  asm volatile("s_wait_asynccnt %0" :: "n"(N) : "memory");
}

// ---- branch-free exact GELU: 0.5*x*(1+erf(x/sqrt(2))) ----------------------
// erf via Abramowitz-Stegun 7.1.26 (|err| < 1.5e-7), v_rcp_f32 + v_exp_f32.
__device__ __forceinline__ float gelu_exact(float x) {
  const float kInvSqrt2 = 0.70710678118654752440f;
  float z  = x * kInvSqrt2;
  float az = fabsf(z);
  float t  = __builtin_amdgcn_rcpf(fmaf(0.3275911f, az, 1.0f));
  float p  = fmaf(t, 1.061405429f, -1.453152027f);
  p = fmaf(t, p, 1.421413741f);
  p = fmaf(t, p, -0.284496736f);
  p = fmaf(t, p, 0.254829592f);
  p = p * t;
  float e      = __expf(-az * az);          // v_exp_f32
  float erf_az = fmaf(-p, e, 1.0f);
  float erf_z  = copysignf(erf_az, z);
  return 0.5f * x * (1.0f + erf_z);
}

// One wave32 per (batch, f-half): 8192 independent scan chains total.
__global__ __launch_bounds__(HALF, 1) void ddi_scan_kernel(
    const float* __restrict__ x, const float* __restrict__ w,
    const float* __restrict__ bias, float* __restrict__ out) {
  __shared__ float smem[NSTAGE * SLOT_F];         // 15 KB ring, 256B row stride

  const int lane = threadIdx.x;                   // 0..31  == f within half
  const int b    = blockIdx.x >> 1;
  const int half = blockIdx.x & 1;
  const int f    = half * HALF + lane;

  const float* xb = x   + (size_t)b * SEQLEN * FDIM + f;
  float*       ob = out + (size_t)b * SEQLEN * FDIM + f;

  // Chunk t row p: global xb[(PATCH + 3t + p)*FDIM]; per-lane LDS mirror at
  // slot*768 + p*256 + f*4 (same intra-slot layout as global -> shared offsets).
  const uint64_t gbase = (uint64_t)(uintptr_t)(xb + (size_t)PATCH * FDIM);
  const uint32_t lbase = (uint32_t)(uintptr_t)(&smem[0]) + (uint32_t)f * 4u;

  // ---- prologue: fill NSTAGE pipeline slots --------------------------------
#pragma unroll
  for (int t = 0; t < NSTAGE; ++t) {
    async_stage(lbase + (uint32_t)t * SLOT_B, gbase + (uint64_t)t * SLOT_B);
  }

  // ---- uniform weights / bias (overlaps with async fills) ------------------
  const float w00 = w[0], w01 = w[1], w02 = w[2];
  const float w10 = w[3], w11 = w[4], w12 = w[5];
  const float w20 = w[6], w21 = w[7], w22 = w[8];
  const float b0 = bias[0], b1 = bias[1], b2 = bias[2];

  // ---- head: copy through + initialize scan state --------------------------
  float s0 = xb[0 * FDIM], s1 = xb[1 * FDIM], s2 = xb[2 * FDIM];
  ob[0 * FDIM] = s0; ob[1 * FDIM] = s1; ob[2 * FDIM] = s2;

  float* optr = ob + (size_t)PATCH * FDIM;

  int      st      = 0;                                   // uniform slot index (SALU)
  uint64_t grefill = gbase + (uint64_t)NSTAGE * SLOT_B;   // chunk t+NSTAGE

  for (int t = 0; t < NCHUNK; ++t) {
    // Async loads retire in order; <= 3*(NSTAGE-1) outstanding => oldest slot done.
    wait_asynccnt_le<3 * (NSTAGE - 1)>();

    const int sb = st * SLOT_F + f;        // uniform*192 + lane: float index
    float c0 = smem[sb];
    float c1 = smem[sb + FDIM];
    float c2 = smem[sb + 2 * FDIM];

    // agg[p] = sum_h W[p,h]*state[h] + bias[p]   (einsum 'bhf,ph->bpf')
    float a0 = fmaf(w00, s0, fmaf(w01, s1, fmaf(w02, s2, b0)));
    float a1 = fmaf(w10, s0, fmaf(w11, s1, fmaf(w12, s2, b1)));
    float a2 = fmaf(w20, s0, fmaf(w21, s1, fmaf(w22, s2, b2)));

    a0 = gelu_exact(a0);
    a1 = gelu_exact(a1);
    a2 = gelu_exact(a2);

    s0 = a0 + c0;
    s1 = a1 + c1;
    s2 = a2 + c2;

    optr[0]        = s0;
    optr[FDIM]     = s1;
    optr[2 * FDIM] = s2;
    optr += PATCH * FDIM;

    // Refill the slot just consumed with chunk t+NSTAGE (same slot index).
    if (t + NSTAGE < NCHUNK) {
      async_stage(lbase + (uint32_t)st * SLOT_B, grefill);
    }
    grefill += SLOT_B;
    st = (st == NSTAGE - 1) ? 0 : st + 1;   // uniform -> s_cselect, off VALU path
  }
  // S_ENDPGM performs an implicit wait-idle; nothing left to drain explicitly.
}

extern "C" void kernel_launch(void* const* d_in, const int* in_sizes, int n_in,
                              void* d_out, int out_size, void* d_ws, size_t ws_size,
                              hipStream_t stream) {
  (void)in_sizes; (void)n_in; (void)out_size; (void)d_ws; (void)ws_size;
  const float* x    = (const float*)d_in[0];
  const float* w    = (const float*)d_in[1];   // (3,3) row-major
  const float* bias = (const float*)d_in[2];   // (3,)
  float* out        = (float*)d_out;           // (128, 3000, 64)

  dim3 grid(BATCH * (FDIM / HALF));   // 256 single-wave workgroups
  dim3 block(HALF);                   // one wave32 each
  ddi_scan_kernel<<<grid, block, 0, stream>>>(x, w, bias, out);
}